// GCN_16252156248489
// MI455X (gfx1250) — compile-verified
//
#include <hip/hip_runtime.h>
#include <math.h>

typedef __attribute__((ext_vector_type(2))) float v2f;
typedef __attribute__((ext_vector_type(8))) float v8f;

#define THREADS 256

// ---------------------------------------------------------------- utilities
__global__ __launch_bounds__(THREADS)
void zero_kernel(float* __restrict__ p, size_t n) {
    size_t i = (size_t)blockIdx.x * THREADS + threadIdx.x;
    if (i < n) p[i] = 0.0f;
}

// one thread per edge: accumulate degrees
__global__ __launch_bounds__(THREADS)
void degree_kernel(const int* __restrict__ src, const int* __restrict__ dst,
                   float* __restrict__ deg_out, float* __restrict__ deg_in, int E) {
    int e = blockIdx.x * THREADS + threadIdx.x;
    if (e >= E) return;
    __hip_atomic_fetch_add(&deg_out[src[e]], 1.0f, __ATOMIC_RELAXED, __HIP_MEMORY_SCOPE_AGENT);
    __hip_atomic_fetch_add(&deg_in[dst[e]],  1.0f, __ATOMIC_RELAXED, __HIP_MEMORY_SCOPE_AGENT);
}

// deg -> clamp(deg,1)^{-1/2}, in place for both arrays
__global__ __launch_bounds__(THREADS)
void norm_kernel(float* __restrict__ a, float* __restrict__ b, int N) {
    int i = blockIdx.x * THREADS + threadIdx.x;
    if (i >= N) return;
    float x = a[i]; x = x < 1.0f ? 1.0f : x; a[i] = 1.0f / sqrtf(x);
    float y = b[i]; y = y < 1.0f ? 1.0f : y; b[i] = 1.0f / sqrtf(y);
}

// ---------------------------------------------------------------- layer-1 aggregation
// one wave per edge, 32 lanes x float4 = 128 features
__global__ __launch_bounds__(THREADS)
void agg1_kernel(const float* __restrict__ h, const int* __restrict__ src,
                 const int* __restrict__ dst, const float* __restrict__ norm_out,
                 float* __restrict__ agg, int E) {
    int wave = (blockIdx.x * THREADS + threadIdx.x) >> 5;
    int lane = threadIdx.x & 31;
    if (wave >= E) return;
    int s = src[wave], d = dst[wave];
    float w = norm_out[s];
    const float4* hp = (const float4*)(h + (size_t)s * 128);
    float4 v = hp[lane];
    float* o = agg + (size_t)d * 128 + lane * 4;
    __hip_atomic_fetch_add(o + 0, v.x * w, __ATOMIC_RELAXED, __HIP_MEMORY_SCOPE_AGENT);
    __hip_atomic_fetch_add(o + 1, v.y * w, __ATOMIC_RELAXED, __HIP_MEMORY_SCOPE_AGENT);
    __hip_atomic_fetch_add(o + 2, v.z * w, __ATOMIC_RELAXED, __HIP_MEMORY_SCOPE_AGENT);
    __hip_atomic_fetch_add(o + 3, v.w * w, __ATOMIC_RELAXED, __HIP_MEMORY_SCOPE_AGENT);
}

// ---------------------------------------------------------------- GEMM1 (WMMA f32)
// H1 = relu( (agg1 @ W1) * norm_in[:,None] + b1 )   [N x 128] @ [128 x 256]
// one wave per 16x64 output strip (4 accumulator tiles, A fragment reused)
__global__ __launch_bounds__(THREADS)
void gemm1_kernel(const float* __restrict__ A, const float* __restrict__ W,
                  const float* __restrict__ bias, const float* __restrict__ nin,
                  float* __restrict__ out, int Mtiles, int N) {
    int wid  = (blockIdx.x * THREADS + threadIdx.x) >> 5;
    int lane = threadIdx.x & 31;
    int tileM = wid >> 2;          // 4 n-groups of 64 cols
    int ng    = wid & 3;
    if (tileM >= Mtiles) return;

    int half = lane >> 4;          // 0 or 1
    int ln   = lane & 15;          // A row / B col / D col
    int koff = half * 2;

    int rowA = tileM * 16 + ln;
    if (rowA > N - 1) rowA = N - 1;      // clamp; garbage rows never stored
    const float* arow  = A + (size_t)rowA * 128 + koff;
    const float* wbase = W + (size_t)koff * 256 + ng * 64 + ln;

    v8f acc[4] = {};
    for (int k = 0; k < 128; k += 4) {
        v2f a; a.x = arow[k]; a.y = arow[k + 1];
        const float* wk = wbase + (size_t)k * 256;
#pragma unroll
        for (int t = 0; t < 4; ++t) {
            v2f b; b.x = wk[t * 16]; b.y = wk[t * 16 + 256];
            acc[t] = __builtin_amdgcn_wmma_f32_16x16x4_f32(
                false, a, false, b, (short)0, acc[t], false, false);
        }
    }
#pragma unroll
    for (int t = 0; t < 4; ++t) {
        int col = ng * 64 + t * 16 + ln;
        float bv = bias[col];
#pragma unroll
        for (int r = 0; r < 8; ++r) {
            int row = tileM * 16 + half * 8 + r;
            if (row < N) {
                float v = acc[t][r] * nin[row] + bv;
                out[(size_t)row * 256 + col] = v > 0.0f ? v : 0.0f;
            }
        }
    }
}

// ---------------------------------------------------------------- GEMM2 (WMMA f32)
// tmp2 = (H1 @ W2) * norm_out[:,None]   [N x 256] @ [256 x 64]
// one wave per 16-row tile, computes all 64 output cols
__global__ __launch_bounds__(THREADS)
void gemm2_kernel(const float* __restrict__ A, const float* __restrict__ W,
                  const float* __restrict__ nout, float* __restrict__ out,
                  int Mtiles, int N) {
    int wid  = (blockIdx.x * THREADS + threadIdx.x) >> 5;
    int lane = threadIdx.x & 31;
    if (wid >= Mtiles) return;
    int tileM = wid;

    int half = lane >> 4;
    int ln   = lane & 15;
    int koff = half * 2;

    int rowA = tileM * 16 + ln;
    if (rowA > N - 1) rowA = N - 1;
    const float* arow  = A + (size_t)rowA * 256 + koff;
    const float* wbase = W + (size_t)koff * 64 + ln;

    v8f acc[4] = {};
    for (int k = 0; k < 256; k += 4) {
        v2f a; a.x = arow[k]; a.y = arow[k + 1];
        const float* wk = wbase + (size_t)k * 64;
#pragma unroll
        for (int t = 0; t < 4; ++t) {
            v2f b; b.x = wk[t * 16]; b.y = wk[t * 16 + 64];
            acc[t] = __builtin_amdgcn_wmma_f32_16x16x4_f32(
                false, a, false, b, (short)0, acc[t], false, false);
        }
    }
#pragma unroll
    for (int t = 0; t < 4; ++t) {
        int col = t * 16 + ln;
#pragma unroll
        for (int r = 0; r < 8; ++r) {
            int row = tileM * 16 + half * 8 + r;
            if (row < N)
                out[(size_t)row * 64 + col] = acc[t][r] * nout[row];
        }
    }
}

// ---------------------------------------------------------------- layer-2 aggregation
// one wave per edge, 32 lanes x float2 = 64 features, accumulate into d_out
__global__ __launch_bounds__(THREADS)
void agg2_kernel(const float* __restrict__ P, const int* __restrict__ src,
                 const int* __restrict__ dst, float* __restrict__ out, int E) {
    int wave = (blockIdx.x * THREADS + threadIdx.x) >> 5;
    int lane = threadIdx.x & 31;
    if (wave >= E) return;
    int s = src[wave], d = dst[wave];
    const float2* p = (const float2*)(P + (size_t)s * 64);
    float2 v = p[lane];
    float* o = out + (size_t)d * 64 + lane * 2;
    __hip_atomic_fetch_add(o + 0, v.x, __ATOMIC_RELAXED, __HIP_MEMORY_SCOPE_AGENT);
    __hip_atomic_fetch_add(o + 1, v.y, __ATOMIC_RELAXED, __HIP_MEMORY_SCOPE_AGENT);
}

// out = out * norm_in[:,None] + b2  (in place)
__global__ __launch_bounds__(THREADS)
void final_kernel(float* __restrict__ out, const float* __restrict__ nin,
                  const float* __restrict__ b2, int N) {
    size_t idx = (size_t)blockIdx.x * THREADS + threadIdx.x;
    if (idx >= (size_t)N * 64) return;
    int row = (int)(idx >> 6);
    int c   = (int)(idx & 63);
    out[idx] = out[idx] * nin[row] + b2[c];
}

// ---------------------------------------------------------------- host
extern "C" void kernel_launch(void* const* d_in, const int* in_sizes, int n_in,
                              void* d_out, int out_size, void* d_ws, size_t ws_size,
                              hipStream_t stream) {
    const float* h   = (const float*)d_in[0];
    const int*   src = (const int*)d_in[1];
    const int*   dst = (const int*)d_in[2];
    const float* W1  = (const float*)d_in[3];
    const float* b1  = (const float*)d_in[4];
    const float* W2  = (const float*)d_in[5];
    const float* b2  = (const float*)d_in[6];
    float* out = (float*)d_out;

    const int N = in_sizes[0] / 128;
    const int E = in_sizes[1];
    const int Mtiles = (N + 15) / 16;

    float* ws = (float*)d_ws;
    size_t Na = (size_t)((N + 63) & ~63);
    float* norm_out = ws;                      // N  (deg_out -> norm_out)
    float* norm_in  = ws + Na;                 // N  (deg_in  -> norm_in)
    float* agg1     = ws + 2 * Na;             // N x 128
    float* H1       = agg1 + (size_t)N * 128;  // N x 256
    float* tmp2     = H1 + (size_t)N * 256;    // N x 64

    size_t z1 = 2 * Na + (size_t)N * 128;
    zero_kernel<<<(unsigned)((z1 + THREADS - 1) / THREADS), THREADS, 0, stream>>>(ws, z1);
    size_t z2 = (size_t)N * 64;
    zero_kernel<<<(unsigned)((z2 + THREADS - 1) / THREADS), THREADS, 0, stream>>>(out, z2);

    degree_kernel<<<(E + THREADS - 1) / THREADS, THREADS, 0, stream>>>(
        src, dst, norm_out, norm_in, E);
    norm_kernel<<<(N + THREADS - 1) / THREADS, THREADS, 0, stream>>>(
        norm_out, norm_in, N);

    agg1_kernel<<<(E + 7) / 8, THREADS, 0, stream>>>(h, src, dst, norm_out, agg1, E);

    gemm1_kernel<<<(Mtiles * 4 + 7) / 8, THREADS, 0, stream>>>(
        agg1, W1, b1, norm_in, H1, Mtiles, N);

    gemm2_kernel<<<(Mtiles + 7) / 8, THREADS, 0, stream>>>(
        H1, W2, norm_out, tmp2, Mtiles, N);

    agg2_kernel<<<(E + 7) / 8, THREADS, 0, stream>>>(tmp2, src, dst, out, E);

    final_kernel<<<(unsigned)((z2 + THREADS - 1) / THREADS), THREADS, 0, stream>>>(
        out, norm_in, b2, N);
}